// YOLOV7_43379169690123
// MI455X (gfx1250) — compile-verified
//
#include <hip/hip_runtime.h>

#define HW     6400          // 80*80
#define CCH    80            // classes
#define NSLOT  64            // striped atomic slots
#define EPSF   1e-7f

typedef float v8f __attribute__((ext_vector_type(8)));
typedef float v2f __attribute__((ext_vector_type(2)));
typedef unsigned int v4u __attribute__((ext_vector_type(4)));
typedef int v8i __attribute__((ext_vector_type(8)));
typedef int v4i __attribute__((ext_vector_type(4)));

// ---- full-precision 32-lane sum using V_WMMA_F32_16X16X4_F32 (B = ones) ----
// With B all-ones, D[m][n] = p_m + p_{m+16}. Lanes 0-15 hold rows 0-7, lanes
// 16-31 hold rows 8-15; 8 adds + one SWAPX16 ds_swizzle completes the sum.
__device__ __forceinline__ float wave_sum32(float p) {
    v2f a; a[0] = p;    a[1] = 0.0f;
    v2f b; b[0] = 1.0f; b[1] = 1.0f;
    v8f c = {0.f, 0.f, 0.f, 0.f, 0.f, 0.f, 0.f, 0.f};
    c = __builtin_amdgcn_wmma_f32_16x16x4_f32(false, a, false, b,
                                              (short)0, c, false, false);
    float s = c[0] + c[1] + c[2] + c[3] + c[4] + c[5] + c[6] + c[7];
    // group-of-32 swizzle, xor=0x10, and=0x1f  ->  lane ^ 16
    int o = __builtin_amdgcn_ds_swizzle(__float_as_int(s), 0x401F);
    return s + __int_as_float(o);
}

__device__ __forceinline__ float bce(float z, float t) {
    return fmaxf(z, 0.0f) - z * t + __logf(1.0f + __expf(-fabsf(z)));
}

// ---------------- kernel 1: zero the striped workspace ----------------
__global__ void yolo_zero_ws(float* ws) {
    int t = blockIdx.x * blockDim.x + threadIdx.x;
    if (t < NSLOT * 5) ws[t] = 0.f;
}

// ---------------- kernel 2: xy / wh / ciou / conf losses ----------------
__global__ __launch_bounds__(256)
void yolo_box(const float* __restrict__ in,  const float* __restrict__ mask,
              const float* __restrict__ obj, const float* __restrict__ tx,
              const float* __restrict__ ty,  const float* __restrict__ tw,
              const float* __restrict__ th,  const float* __restrict__ tsc,
              float* __restrict__ ws) {
    __shared__ float red[8][4];
    int idx  = blockIdx.x * 256 + threadIdx.x;     // [0, 307200), exact
    int slab = idx / HW;                           // b*3 + a
    int hw   = idx - slab * HW;

    const float* p = in + (size_t)slab * 85 * HW + hw;   // coalesced per plane
    float zx = __builtin_nontemporal_load(p);
    float zy = __builtin_nontemporal_load(p + HW);
    float wv = __builtin_nontemporal_load(p + 2 * HW);
    float hv = __builtin_nontemporal_load(p + 3 * HW);
    float zc = __builtin_nontemporal_load(p + 4 * HW);

    float mk = mask[idx], ob = obj[idx];
    float gx = tx[idx], gy = ty[idx], gw = tw[idx], gh = th[idx], sc = tsc[idx];

    float x = 1.0f / (1.0f + __expf(-zx));
    float y = 1.0f / (1.0f + __expf(-zy));

    float p_conf = ob * bce(zc, mk);
    float p_xy   = sc * (fabsf(x - gx) + fabsf(y - gy));
    float p_wh   = sc * (fabsf(wv - gw) + fabsf(hv - gh));

    // CIoU (forward value; alpha is stop_gradient'd so value = v/(v-iou+1+eps))
    float px1 = x - wv * 0.5f, px2 = x + wv * 0.5f;
    float py1 = y - hv * 0.5f, py2 = y + hv * 0.5f;
    float gx1 = gx - gw * 0.5f, gx2 = gx + gw * 0.5f;
    float gy1 = gy - gh * 0.5f, gy2 = gy + gh * 0.5f;
    float iw = fmaxf(fminf(px2, gx2) - fmaxf(px1, gx1), 0.f);
    float ih = fmaxf(fminf(py2, gy2) - fmaxf(py1, gy1), 0.f);
    float inter = iw * ih;
    float uni   = wv * hv + gw * gh - inter + EPSF;
    float iou   = inter / uni;
    float cw = fmaxf(px2, gx2) - fminf(px1, gx1);
    float ch = fmaxf(py2, gy2) - fminf(py1, gy1);
    float c2 = cw * cw + ch * ch + EPSF;
    float rho2 = (x - gx) * (x - gx) + (y - gy) * (y - gy);
    float da = atanf(gw / (gh + EPSF)) - atanf(wv / (hv + EPSF));
    float v  = 0.405284734569f * da * da;          // 4/pi^2
    float alpha = v / (v - iou + 1.0f + EPSF);
    float lbox  = 1.0f - (iou - (rho2 / c2 + v * alpha));
    float p_iou = mk * sc * lbox;

    // 4 WMMA wave reductions, single barrier, 4 striped atomics
    float s0 = wave_sum32(p_xy);
    float s1 = wave_sum32(p_wh);
    float s2 = wave_sum32(p_iou);
    float s3 = wave_sum32(p_conf);
    int lane = threadIdx.x & 31, w = threadIdx.x >> 5;
    if (lane == 0) { red[w][0] = s0; red[w][1] = s1; red[w][2] = s2; red[w][3] = s3; }
    __syncthreads();
    if (threadIdx.x == 0) {
        float r0 = 0.f, r1 = 0.f, r2 = 0.f, r3 = 0.f;
        #pragma unroll
        for (int i = 0; i < 8; ++i) {
            r0 += red[i][0]; r1 += red[i][1]; r2 += red[i][2]; r3 += red[i][3];
        }
        int slot = (blockIdx.x & (NSLOT - 1)) * 5;
        atomicAdd(&ws[slot + 0], r0);
        atomicAdd(&ws[slot + 1], r1);
        atomicAdd(&ws[slot + 2], r2);
        atomicAdd(&ws[slot + 3], r3);
    }
}

// ---------------- kernel 3: class BCE loss (dominant stream) ----------------
// One block per (slab, 64-hw) tile. The 64x80 tcls tile (20 KB, contiguous in
// global) is staged into LDS by ONE Tensor-Data-Mover descriptor with hardware
// padding: pad_interval=16 DWORDs, pad_amount=1 DWORD  =>  element n lands at
// LDS dword n + n/16 (row stride 85). Transposed reads (fixed c, hw across
// lanes) hit banks hw*21 mod 64 -> conflict-free. Input class planes are then
// read fully coalesced.
__global__ __launch_bounds__(256)
void yolo_cls(const float* __restrict__ in, const float* __restrict__ mask,
              const float* __restrict__ tcls, float* __restrict__ ws) {
    __shared__ float tile[64 * 85];                // 5120 data + 320 pad dwords
    __shared__ float red[8];
    int t       = threadIdx.x;
    int slab    = blockIdx.x / 100;                // 100 tiles per slab
    int base_hw = (blockIdx.x - slab * 100) << 6;

    if (t < 32) {                                  // wave 0 drives the TDM
        unsigned ldsb = (unsigned)(size_t)(&tile[0]);
        unsigned long long gaddr = (unsigned long long)tcls
            + (unsigned long long)(((unsigned)slab * (unsigned)HW + (unsigned)base_hw)
                                   * (unsigned)CCH) * 4ull;
        v4u g0;
        g0[0] = 1u;                                          // count=1, no gather
        g0[1] = ldsb;                                        // lds_addr
        g0[2] = (unsigned)(gaddr & 0xFFFFFFFFull);           // global_addr[31:0]
        g0[3] = (unsigned)((gaddr >> 32) & 0x1FFFFFFull)     // global_addr[56:32]
              | (2u << 30);                                  // type=2 (image)
        v8i g1;
        g1[0] = (2 << 16)        // data_size = 4B
              | (1 << 20)        // pad_enable
              | (3 << 22)        // pad_interval: 16 DWORDs
              | (0 << 25);       // pad_amount:   1 DWORD
        g1[1] = (int)((5120u & 0xFFFFu) << 16);  // tensor_dim0[15:0]
        g1[2] = 0;                               // tensor_dim0[31:16], tensor_dim1 lo
        g1[3] = (int)(5120u << 16);              // tile_dim0 = 5120 (1D transfer)
        g1[4] = 0;                               // tile_dim1/2 unused
        g1[5] = 5120;                            // tensor_dim0_stride lo
        g1[6] = 0;
        g1[7] = 0;
        v4i z4 = {0, 0, 0, 0};
        v8i z8 = {0, 0, 0, 0, 0, 0, 0, 0};
        __builtin_amdgcn_tensor_load_to_lds(g0, g1, z4, z4, z8, 0);
        __builtin_amdgcn_s_wait_tensorcnt(0);
    }
    __syncthreads();

    int hwl = t & 63;
    int c0  = t >> 6;
    float m = mask[(size_t)slab * HW + base_hw + hwl];
    const float* pin = in + (size_t)slab * 85 * HW + (size_t)5 * HW + base_hw + hwl;

    float acc = 0.f;
    #pragma unroll 4
    for (int k = 0; k < 20; ++k) {
        int c   = c0 + (k << 2);
        float z  = __builtin_nontemporal_load(pin + (size_t)c * HW); // coalesced
        float tv = tile[hwl * 85 + c + (c >> 4)];                    // bank-safe
        acc += bce(z, tv);
    }
    acc *= m;

    float s = wave_sum32(acc);
    int lane = t & 31, w = t >> 5;
    if (lane == 0) red[w] = s;
    __syncthreads();
    if (t == 0) {
        float r = 0.f;
        #pragma unroll
        for (int i = 0; i < 8; ++i) r += red[i];
        atomicAdd(&ws[(blockIdx.x & (NSLOT - 1)) * 5 + 4], r);
    }
}

// ---------------- kernel 4: fold slots, apply scales ----------------
__global__ void yolo_finalize(const float* __restrict__ ws, float* __restrict__ out) {
    int t = threadIdx.x;
    if (t < 5) {
        float s = 0.f;
        for (int j = 0; j < NSLOT; ++j) s += ws[j * 5 + t];
        // [xy/bs^2, wh/bs^2, iou/bs, conf/bs, cls/bs], bs = 16
        const float sc[5] = {1.f/256.f, 1.f/256.f, 1.f/16.f, 1.f/16.f, 1.f/16.f};
        out[t] = s * sc[t];
    }
}

extern "C" void kernel_launch(void* const* d_in, const int* in_sizes, int n_in,
                              void* d_out, int out_size, void* d_ws, size_t ws_size,
                              hipStream_t stream) {
    const float* in   = (const float*)d_in[0];
    const float* mask = (const float*)d_in[1];
    const float* obj  = (const float*)d_in[2];
    const float* tx   = (const float*)d_in[3];
    const float* ty   = (const float*)d_in[4];
    const float* tw   = (const float*)d_in[5];
    const float* th   = (const float*)d_in[6];
    const float* tsc  = (const float*)d_in[7];
    const float* tcls = (const float*)d_in[8];
    float* out = (float*)d_out;
    float* ws  = (float*)d_ws;

    yolo_zero_ws<<<1, NSLOT * 5, 0, stream>>>(ws);
    yolo_box<<<307200 / 256, 256, 0, stream>>>(in, mask, obj, tx, ty, tw, th, tsc, ws);
    yolo_cls<<<48 * 100, 256, 0, stream>>>(in, mask, tcls, ws);
    yolo_finalize<<<1, 32, 0, stream>>>(ws, out);
}